// CrossLevelAttention_85873576117156
// MI455X (gfx1250) — compile-verified
//
#include <hip/hip_runtime.h>
#include <hip/hip_bf16.h>

#define HIDDEN 128
#define HEADS  8

typedef float v2f __attribute__((ext_vector_type(2)));
typedef float v8f __attribute__((ext_vector_type(8)));

// ---------------- small bookkeeping kernels ----------------

__global__ void init_counts(int* counts, int C) {
    int i = blockIdx.x * blockDim.x + threadIdx.x;
    if (i < C) counts[i] = 0;
}

__global__ void count_cells(const int* __restrict__ labels, int* __restrict__ counts,
                            int N, int C) {
    int i = blockIdx.x * blockDim.x + threadIdx.x;
    if (i < N) {
        int l = labels[i];
        if (l >= 0 && l < C) atomicAdd(&counts[l], 1);
    }
}

__global__ void scan_counts(const int* __restrict__ counts, int* __restrict__ starts,
                            int* __restrict__ cursor, int C) {
    if (blockIdx.x == 0 && threadIdx.x == 0) {
        int acc = 0;
        for (int c = 0; c < C; ++c) { starts[c] = acc; cursor[c] = acc; acc += counts[c]; }
        starts[C] = acc;
    }
}

__global__ void scatter_cells_idx(const int* __restrict__ labels, int* __restrict__ cursor,
                                  int* __restrict__ mlist, int N, int C) {
    int i = blockIdx.x * blockDim.x + threadIdx.x;
    if (i < N) {
        int l = labels[i];
        if (l >= 0 && l < C) {
            int p = atomicAdd(&cursor[l], 1);
            mlist[p] = i;
        }
    }
}

// q = (tissue @ Wq^T + bq) * 1/sqrt(16)   -- tiny (2000x128), plain VALU
__global__ void qproj_kernel(const float* __restrict__ tissue, const float* __restrict__ bu_in_w,
                             const float* __restrict__ bu_in_b, float* __restrict__ qbuf, int C) {
    int t = blockIdx.x * blockDim.x + threadIdx.x;
    if (t < C * HIDDEN) {
        int c = t >> 7, o = t & 127;
        const float* x = tissue + (size_t)c * HIDDEN;
        const float* w = bu_in_w + (size_t)o * HIDDEN;   // Wq = rows 0..127
        float acc = bu_in_b[o];
        #pragma unroll 8
        for (int k = 0; k < HIDDEN; ++k) acc = fmaf(x[k], w[k], acc);
        qbuf[t] = acc * 0.25f;                            // scale = 1/sqrt(d), d=16
    }
}

// ---------------- main fused bottom-up attention: one workgroup per cluster ----------------
// Wk/Wv staged in LDS (stride-132 padding -> conflict-free ds_load_b64 fragments).
// 8 waves; each wave owns 32-cell pair-chunks: two A tiles share every B fragment.
// Online softmax kept entirely in registers; cross-wave merge in LDS at the end.

#define LWSTRIDE 132   // padded LDS row stride (floats); bank = 4*n16+2*hi+4*kk, conflict-free

__global__ __launch_bounds__(256)
void cluster_attn_kernel(const float* __restrict__ cell,
                         const float* __restrict__ tissue,
                         const int*   __restrict__ mlist,
                         const int*   __restrict__ starts,
                         const float* __restrict__ qbuf,
                         const float* __restrict__ bu_in_w,
                         const float* __restrict__ bu_in_b,
                         const float* __restrict__ bu_out_w,
                         const float* __restrict__ bu_out_b,
                         const float* __restrict__ td_in_w,
                         const float* __restrict__ td_in_b,
                         const float* __restrict__ td_out_w,
                         const float* __restrict__ td_out_b,
                         float* __restrict__ tissue_out,
                         float* __restrict__ tout,
                         int C)
{
    __shared__ __align__(16) float lw[256 * LWSTRIDE];  // Wk rows 0..127, Wv rows 128..255
    __shared__ float smax_l[8][HEADS];
    __shared__ float sden_l[8][HEADS];
    __shared__ float snum_l[8][HEADS][16];
    __shared__ float attn_l[HIDDEN];
    __shared__ float row_l[HIDDEN];
    __shared__ float u_l[HIDDEN];

    const int c    = blockIdx.x;
    const int tid  = threadIdx.x;
    const int wave = tid >> 5;
    const int lane = tid & 31;
    const int n16  = lane & 15;   // N (feature) index inside 16-wide tile / j index
    const int hi   = lane >> 4;   // half-wave select (K split for A/B fragments)

    // ---- stage Wk|Wv (global rows 128..383 of bu_in_w) into padded LDS, one row per thread
    {
        const float4* src = (const float4*)(bu_in_w + 128 * 128 + (size_t)tid * HIDDEN);
        float* dst = lw + (size_t)tid * LWSTRIDE;
        #pragma unroll
        for (int j = 0; j < 32; ++j) *(float4*)(dst + 4 * j) = src[j];
    }
    __syncthreads();

    const int s0  = starts[c];
    const int cnt = starts[c + 1] - s0;

    const float* bk = bu_in_b + 128;
    const float* bv = bu_in_b + 256;

    // per-lane scaled q values: q[c, h, n16]
    float qreg[HEADS];
    #pragma unroll
    for (int h = 0; h < HEADS; ++h) qreg[h] = qbuf[c * HIDDEN + h * 16 + n16];

    float rmax[HEADS], rden[HEADS], rnum[HEADS];
    #pragma unroll
    for (int h = 0; h < HEADS; ++h) { rmax[h] = -3.0e38f; rden[h] = 0.0f; rnum[h] = 0.0f; }

    const int npair = (cnt + 31) >> 5;
    for (int pp = wave; pp < npair; pp += 8) {
        const int base0 = pp << 5;
        const int base1 = base0 + 16;

        // prefetch next pair's scattered member rows (compiler can't see through mlist)
        const int np = pp + 8;
        if (np < npair) {
            const int q0 = np * 32 + n16;
            const int q1 = q0 + 16;
            if (q0 < cnt) __builtin_prefetch(cell + (size_t)mlist[s0 + q0] * HIDDEN + 64 * hi, 0, 0);
            if (q1 < cnt) __builtin_prefetch(cell + (size_t)mlist[s0 + q1] * HIDDEN + 64 * hi, 0, 0);
        }

        const int  m0  = base0 + n16;
        const int  m1  = base1 + n16;
        const bool ok0 = m0 < cnt;
        const bool ok1 = m1 < cnt;
        const float* x0 = cell + (size_t)(ok0 ? mlist[s0 + m0] : 0) * HIDDEN + 2 * hi;
        const float* x1 = cell + (size_t)(ok1 ? mlist[s0 + m1] : 0) * HIDDEN + 2 * hi;

        // A fragments (16x4 f32 WMMA A layout), two 16-cell tiles
        v2f A0[32], A1[32];
        #pragma unroll
        for (int kk = 0; kk < 32; ++kk) {
            v2f a0 = *(const v2f*)(x0 + 4 * kk);
            v2f a1 = *(const v2f*)(x1 + 4 * kk);
            if (!ok0) { a0.x = 0.0f; a0.y = 0.0f; }
            if (!ok1) { a1.x = 0.0f; a1.y = 0.0f; }
            A0[kk] = a0; A1[kk] = a1;
        }

        #pragma unroll 1
        for (int h = 0; h < HEADS; ++h) {
            const float* wkl = lw + (size_t)(h * 16 + n16) * LWSTRIDE + 2 * hi;          // Wk row
            const float* wvl = lw + (size_t)(128 + h * 16 + n16) * LWSTRIDE + 2 * hi;    // Wv row

            // ---- K projection: both tiles share every B fragment (ds_load_b64)
            v8f k0 = {0.f,0.f,0.f,0.f,0.f,0.f,0.f,0.f};
            v8f k1 = {0.f,0.f,0.f,0.f,0.f,0.f,0.f,0.f};
            #pragma unroll
            for (int kk = 0; kk < 32; ++kk) {
                v2f b = *(const v2f*)(wkl + 4 * kk);
                k0 = __builtin_amdgcn_wmma_f32_16x16x4_f32(false, A0[kk], false, b, (short)0, k0, false, false);
                k1 = __builtin_amdgcn_wmma_f32_16x16x4_f32(false, A1[kk], false, b, (short)0, k1, false, false);
            }

            // ---- scores for both tiles: butterfly-reduce over lanes (n dim)
            const float bkv = bk[h * 16 + n16];
            float s0a[8], s1a[8];
            #pragma unroll
            for (int r = 0; r < 8; ++r) {
                s0a[r] = (k0[r] + bkv) * qreg[h];
                s0a[r] += __shfl_xor(s0a[r], 1, 32);
                s0a[r] += __shfl_xor(s0a[r], 2, 32);
                s0a[r] += __shfl_xor(s0a[r], 4, 32);
                s0a[r] += __shfl_xor(s0a[r], 8, 32);
                s1a[r] = (k1[r] + bkv) * qreg[h];
                s1a[r] += __shfl_xor(s1a[r], 1, 32);
                s1a[r] += __shfl_xor(s1a[r], 2, 32);
                s1a[r] += __shfl_xor(s1a[r], 4, 32);
                s1a[r] += __shfl_xor(s1a[r], 8, 32);
            }
            // mask padding rows, chunk max over all 32 rows
            float lmax = -3.0e38f;
            #pragma unroll
            for (int r = 0; r < 8; ++r) {
                const bool o0 = (base0 + r + 8 * hi) < cnt;
                const bool o1 = (base1 + r + 8 * hi) < cnt;
                s0a[r] = o0 ? s0a[r] : -3.0e38f;
                s1a[r] = o1 ? s1a[r] : -3.0e38f;
                lmax = fmaxf(lmax, fmaxf(s0a[r], s1a[r]));
            }
            lmax = fmaxf(lmax, __shfl_xor(lmax, 16, 32));
            const float newmax = fmaxf(rmax[h], lmax);
            const float corr   = __expf(rmax[h] - newmax);
            float e0[8], e1[8]; float dsum = 0.0f;
            #pragma unroll
            for (int r = 0; r < 8; ++r) {
                e0[r] = (s0a[r] > -1.0e38f) ? __expf(s0a[r] - newmax) : 0.0f;
                e1[r] = (s1a[r] > -1.0e38f) ? __expf(s1a[r] - newmax) : 0.0f;
                dsum += e0[r] + e1[r];
            }
            dsum += __shfl_xor(dsum, 16, 32);

            // ---- V projection, same shared B fragments
            v8f v0 = {0.f,0.f,0.f,0.f,0.f,0.f,0.f,0.f};
            v8f v1 = {0.f,0.f,0.f,0.f,0.f,0.f,0.f,0.f};
            #pragma unroll
            for (int kk = 0; kk < 32; ++kk) {
                v2f b = *(const v2f*)(wvl + 4 * kk);
                v0 = __builtin_amdgcn_wmma_f32_16x16x4_f32(false, A0[kk], false, b, (short)0, v0, false, false);
                v1 = __builtin_amdgcn_wmma_f32_16x16x4_f32(false, A1[kk], false, b, (short)0, v1, false, false);
            }
            const float bvv = bv[h * 16 + n16];
            float p = 0.0f;
            #pragma unroll
            for (int r = 0; r < 8; ++r) {
                p = fmaf(e0[r], v0[r] + bvv, p);
                p = fmaf(e1[r], v1[r] + bvv, p);
            }
            p += __shfl_xor(p, 16, 32);    // combine halves -> full sum over m

            rmax[h] = newmax;
            rden[h] = fmaf(rden[h], corr, dsum);
            rnum[h] = fmaf(rnum[h], corr, p);
        }
    }

    // ---- cross-wave merge
    if (lane < 16) {
        #pragma unroll
        for (int h = 0; h < HEADS; ++h) snum_l[wave][h][n16] = rnum[h];
    }
    if (lane == 0) {
        #pragma unroll
        for (int h = 0; h < HEADS; ++h) { smax_l[wave][h] = rmax[h]; sden_l[wave][h] = rden[h]; }
    }
    __syncthreads();

    if (tid < HIDDEN) {
        const int h = tid >> 4, j = tid & 15;
        float gmax = -3.0e38f;
        #pragma unroll
        for (int w = 0; w < 8; ++w) gmax = fmaxf(gmax, smax_l[w][h]);
        float den = 0.0f, num = 0.0f;
        #pragma unroll
        for (int w = 0; w < 8; ++w) {
            const float f = __expf(smax_l[w][h] - gmax);
            den = fmaf(sden_l[w][h], f, den);
            num = fmaf(snum_l[w][h][j], f, num);
        }
        attn_l[tid] = num / (den > 0.0f ? den : 1.0f);
    }
    __syncthreads();

    // bu_out = attn @ bu_out_w^T + b  (or passthrough if cluster empty)
    if (tid < HIDDEN) {
        float o;
        if (cnt > 0) {
            float acc = bu_out_b[tid];
            const float* w = bu_out_w + (size_t)tid * HIDDEN;
            #pragma unroll 8
            for (int k = 0; k < HIDDEN; ++k) acc = fmaf(attn_l[k], w[k], acc);
            o = acc;
        } else {
            o = tissue[(size_t)c * HIDDEN + tid];
        }
        tissue_out[(size_t)c * HIDDEN + tid] = o;
        row_l[tid] = o;
    }
    __syncthreads();

    // top-down: t_out = (tissue_new @ Wv2^T + bv2) @ td_out_w^T + td_out_b
    if (tid < HIDDEN) {
        const float* Wv2 = td_in_w + 2 * 128 * 128;   // rows 256..383
        float acc = td_in_b[256 + tid];
        const float* w = Wv2 + (size_t)tid * HIDDEN;
        #pragma unroll 8
        for (int k = 0; k < HIDDEN; ++k) acc = fmaf(row_l[k], w[k], acc);
        u_l[tid] = acc;
    }
    __syncthreads();
    if (tid < HIDDEN) {
        float acc = td_out_b[tid];
        const float* w = td_out_w + (size_t)tid * HIDDEN;
        #pragma unroll 8
        for (int k = 0; k < HIDDEN; ++k) acc = fmaf(u_l[k], w[k], acc);
        tout[(size_t)c * HIDDEN + tid] = acc;
    }
}

// ---------------- top-down gather: cell_new[i] = valid ? t_out[label[i]] : cell[i] ----------------

__global__ void td_gather_kernel(const float4* __restrict__ cell4,
                                 const int*    __restrict__ labels,
                                 const float4* __restrict__ tout4,
                                 float4* __restrict__ out4, int N, int C) {
    int t = blockIdx.x * blockDim.x + threadIdx.x;
    int i = t >> 5;          // cell index
    int j = t & 31;          // float4 within 128-float row
    if (i < N) {
        int l = labels[i];
        if (l >= 0) {
            int lc = l < C ? l : C - 1;
            out4[(size_t)i * 32 + j] = tout4[(size_t)lc * 32 + j];
        } else {
            out4[(size_t)i * 32 + j] = cell4[(size_t)i * 32 + j];
        }
    }
}

// ---------------- launch ----------------

extern "C" void kernel_launch(void* const* d_in, const int* in_sizes, int n_in,
                              void* d_out, int out_size, void* d_ws, size_t ws_size,
                              hipStream_t stream) {
    const float* cell     = (const float*)d_in[0];
    const float* tissue   = (const float*)d_in[1];
    const int*   labels   = (const int*)d_in[2];
    // d_in[3] tissue_batch unused, d_in[4] num_heads fixed at 8
    const float* bu_in_w  = (const float*)d_in[5];
    const float* bu_in_b  = (const float*)d_in[6];
    const float* bu_out_w = (const float*)d_in[7];
    const float* bu_out_b = (const float*)d_in[8];
    const float* td_in_w  = (const float*)d_in[9];
    const float* td_in_b  = (const float*)d_in[10];
    const float* td_out_w = (const float*)d_in[11];
    const float* td_out_b = (const float*)d_in[12];

    const int N = in_sizes[0] / HIDDEN;
    const int C = in_sizes[1] / HIDDEN;

    // workspace layout (ints then 16B-aligned floats); ~4 MB total
    int* wsi    = (int*)d_ws;
    int* counts = wsi;                 // [C]
    int* starts = wsi + C;             // [C+1]
    int* cursor = wsi + 2 * C + 1;     // [C]
    int* mlist  = wsi + 3 * C + 1;     // [N]
    size_t iofs = (size_t)(3 * C + 1 + N);
    iofs = (iofs + 3) & ~(size_t)3;    // 16B alignment for float4 reads of tout
    float* qbuf = (float*)d_ws + iofs;             // [C*128] scaled q
    float* tout = qbuf + (size_t)C * HIDDEN;       // [C*128] top-down per-cluster output

    float* out_cell   = (float*)d_out;                       // [N*128]
    float* out_tissue = out_cell + (size_t)N * HIDDEN;       // [C*128]

    init_counts      <<<(C + 255) / 256, 256, 0, stream>>>(counts, C);
    count_cells      <<<(N + 255) / 256, 256, 0, stream>>>(labels, counts, N, C);
    scan_counts      <<<1, 32, 0, stream>>>(counts, starts, cursor, C);
    scatter_cells_idx<<<(N + 255) / 256, 256, 0, stream>>>(labels, cursor, mlist, N, C);
    qproj_kernel     <<<(C * HIDDEN + 255) / 256, 256, 0, stream>>>(tissue, bu_in_w, bu_in_b, qbuf, C);
    cluster_attn_kernel<<<C, 256, 0, stream>>>(cell, tissue, mlist, starts, qbuf,
                                               bu_in_w, bu_in_b, bu_out_w, bu_out_b,
                                               td_in_w, td_in_b, td_out_w, td_out_b,
                                               out_tissue, tout, C);
    td_gather_kernel <<<(N * 32 + 255) / 256, 256, 0, stream>>>(
        (const float4*)cell, labels, (const float4*)tout, (float4*)out_cell, N, C);
}